// bi_tempered_logistic_loss_87308095193839
// MI455X (gfx1250) — compile-verified
//
#include <hip/hip_runtime.h>

#define C_CLASSES 1000
#define CHUNKS 8        // 8 x vec4 per lane x 32 lanes = 1024 slots (1000 used)
#define WPB 8           // waves (rows) per block

typedef float v4f __attribute__((ext_vector_type(4)));

__device__ __forceinline__ float rcp_fast(float x)  { return __builtin_amdgcn_rcpf(x); }
__device__ __forceinline__ float log2_fast(float x) { return __builtin_amdgcn_logf(x); }
__device__ __forceinline__ float exp2_fast(float x) { return __builtin_amdgcn_exp2f(x); }

// exp_t(a, 1.2) with w = 1 + u*s (u = -0.2*a0 >= 0):  w^-5
__device__ __forceinline__ float invpow5(float w) {
  float r  = rcp_fast(w);
  float r2 = r * r;
  float r4 = r2 * r2;
  return r4 * r;
}

__device__ __forceinline__ float wave_sum(float v) {
#pragma unroll
  for (int off = 16; off > 0; off >>= 1) v += __shfl_xor(v, off, 32);
  return v;
}
__device__ __forceinline__ float wave_max(float v) {
#pragma unroll
  for (int off = 16; off > 0; off >>= 1) v = fmaxf(v, __shfl_xor(v, off, 32));
  return v;
}

// per-element loss: A(y) + 5*lab(y)*(1 - 1/w) + w^-6 / 1.2   (w = u + z^0.2)
__device__ __forceinline__ float loss_elem(float u, float c, float y,
                                           float LO, float HI,
                                           float A_lo, float A_hi) {
  float w  = u + c;
  float r  = rcp_fast(w);       // = probs^0.2
  float r2 = r * r;
  float r6 = r2 * r2 * r2;      // = probs^1.2
  bool hot = y > 0.5f;
  float lv = hot ? HI : LO;
  float A  = hot ? A_hi : A_lo;
  return fmaf(5.0f * lv, 1.0f - r, fmaf(r6, (1.0f / 1.2f), A));
}

__global__ __launch_bounds__(WPB * 32)
void bi_tempered_loss_kernel(const float* __restrict__ act,
                             const float* __restrict__ lab,
                             float* __restrict__ out, int nrows) {
  __shared__ __align__(16) float s_lab[WPB * CHUNKS * 128];  // 32 KB

  const int lane = threadIdx.x & 31;
  const int wave = threadIdx.x >> 5;
  const int row  = blockIdx.x * WPB + wave;
  if (row >= nrows) return;

  const float* arow = act + (size_t)row * C_CLASSES;
  const unsigned long long lbase =
      (unsigned long long)(const void*)(lab + (size_t)row * C_CLASSES);

  // ---- load activations (streaming: NT hint, data touched exactly once) ----
  v4f a[CHUNKS];
  float mx = -3.0e38f;
#pragma unroll
  for (int k = 0; k < CHUNKS; ++k) {
    const bool valid = (k < CHUNKS - 1) || (lane < 26); // 1000 = 7*128 + 26*4
    if (valid) {
      a[k] = __builtin_nontemporal_load(
                 (const v4f*)(arow + (k * 128 + lane * 4)));
      mx = fmaxf(mx, fmaxf(fmaxf(a[k].x, a[k].y), fmaxf(a[k].z, a[k].w)));
    } else {
      a[k] = v4f{-1e38f, -1e38f, -1e38f, -1e38f};
    }
  }

  // ---- async-stage labels into this wave's LDS slice; overlaps iterations ----
  const unsigned lds_wave =
      (unsigned)(size_t)(&s_lab[0]) + (unsigned)(wave * CHUNKS * 512);
#pragma unroll
  for (int k = 0; k < CHUNKS; ++k) {
    const bool valid = (k < CHUNKS - 1) || (lane < 26);
    if (valid) {
      unsigned voff  = (unsigned)(k * 512 + lane * 16); // byte offset in row
      unsigned laddr = lds_wave + voff;                 // LDS byte address
      asm volatile("global_load_async_to_lds_b128 %0, %1, %2 th:TH_LOAD_NT"
                   :: "v"(laddr), "v"(voff), "s"(lbase)
                   : "memory");
    }
  }

  mx = wave_max(mx);

  // u = -0.2*(act - mu) >= 0 ; padded slots become huge -> contribute exactly 0
#pragma unroll
  for (int k = 0; k < CHUNKS; ++k) {
    a[k].x = 0.2f * (mx - a[k].x);
    a[k].y = 0.2f * (mx - a[k].y);
    a[k].z = 0.2f * (mx - a[k].z);
    a[k].w = 0.2f * (mx - a[k].w);
  }

  // ---- 5 fixed-point iterations + final normalizer (6 sum passes) ----
  float s = 1.0f;   // z^(1-t2), starts at 1
  float z = 0.0f;
#pragma unroll 1
  for (int it = 0; it < 5; ++it) {
    float acc = 0.0f;
#pragma unroll
    for (int k = 0; k < CHUNKS; ++k) {
      acc += invpow5(fmaf(a[k].x, s, 1.0f));
      acc += invpow5(fmaf(a[k].y, s, 1.0f));
      acc += invpow5(fmaf(a[k].z, s, 1.0f));
      acc += invpow5(fmaf(a[k].w, s, 1.0f));
    }
    z = wave_sum(acc);
    s = exp2_fast(-0.2f * log2_fast(z));  // z^(1-t2), per-row scalar pow
  }
  {
    float acc = 0.0f;
#pragma unroll
    for (int k = 0; k < CHUNKS; ++k) {
      acc += invpow5(fmaf(a[k].x, s, 1.0f));
      acc += invpow5(fmaf(a[k].y, s, 1.0f));
      acc += invpow5(fmaf(a[k].z, s, 1.0f));
      acc += invpow5(fmaf(a[k].w, s, 1.0f));
    }
    z = wave_sum(acc);
  }
  // final w = u + z^0.2  (since norm - mu = 5*(z^0.2 - 1), w = 1 - 0.2*(act-norm))
  const float c = exp2_fast(0.2f * log2_fast(z));

  // smoothed-label constants (labels input is raw one-hot {0,1})
  const float LO    = 0.1f / 999.0f;
  const float HI    = (1.0f - (1000.0f / 999.0f) * 0.1f) + 0.1f / 999.0f; // 0.9
  const float LT_lo = 5.0f * (__powf(LO + 1e-10f, 0.2f) - 1.0f);
  const float LT_hi = 5.0f * (__powf(HI + 1e-10f, 0.2f) - 1.0f);
  const float A_lo  = LO * LT_lo - __powf(LO, 1.2f) * (1.0f / 1.2f);
  const float A_hi  = HI * LT_hi - __powf(HI, 1.2f) * (1.0f / 1.2f);

  // labels finished streaming into LDS by now
  asm volatile("s_wait_asynccnt 0" ::: "memory");

  const v4f* slw = (const v4f*)(&s_lab[wave * CHUNKS * 128]);
  float lsum = 0.0f;
#pragma unroll
  for (int k = 0; k < CHUNKS; ++k) {
    const bool valid = (k < CHUNKS - 1) || (lane < 26);
    if (valid) {
      v4f y = slw[k * 32 + lane];
      lsum += loss_elem(a[k].x, c, y.x, LO, HI, A_lo, A_hi);
      lsum += loss_elem(a[k].y, c, y.y, LO, HI, A_lo, A_hi);
      lsum += loss_elem(a[k].z, c, y.z, LO, HI, A_lo, A_hi);
      lsum += loss_elem(a[k].w, c, y.w, LO, HI, A_lo, A_hi);
    }
  }
  lsum = wave_sum(lsum);
  if (lane == 0) __builtin_nontemporal_store(lsum, out + row);
}

extern "C" void kernel_launch(void* const* d_in, const int* in_sizes, int n_in,
                              void* d_out, int out_size, void* d_ws, size_t ws_size,
                              hipStream_t stream) {
  const float* act = (const float*)d_in[0];
  const float* lab = (const float*)d_in[1];
  float* out = (float*)d_out;
  (void)d_ws; (void)ws_size; (void)n_in; (void)out_size;

  const int nrows  = in_sizes[0] / C_CLASSES;          // 65536
  const int blocks = (nrows + WPB - 1) / WPB;          // 8192 blocks x 256 thr
  bi_tempered_loss_kernel<<<blocks, WPB * 32, 0, stream>>>(act, lab, out, nrows);
}